// AttentionModule_60610578481268
// MI455X (gfx1250) — compile-verified
//
#include <hip/hip_runtime.h>

typedef float v2f __attribute__((ext_vector_type(2)));
typedef float v4f __attribute__((ext_vector_type(4)));
typedef float v8f __attribute__((ext_vector_type(8)));

#define K_SLOTS 17
#define C_DIM   256
#define CI      64
#define NEDGE   20
#define EF_K    128

// Padded pitches (+4 floats) -> A-operand bank = (4*row + k) % 64, conflict-free
#define XS_P   260   // xs rows (k-dim 256)
#define WDT_P  260   // W_down^T rows [n][k], k-dim 256
#define EF_P   132   // ef rows (k-dim 128)
#define WCL_P  132   // W_conv natural [n][k], k-dim 128
#define XD_P   68
#define PE_P   68
#define AGG_P  68

// LDS float offsets (all 16B-aligned)
#define OFF_XS   0                        // 17*260  = 4420
#define OFF_XD   (OFF_XS  + 4420)         // 17*68   = 1156
#define OFF_MNS  (OFF_XD  + 1156)         // 5*64    = 320
#define OFF_EF   (OFF_MNS + 320)          // 20*132  = 2640
#define OFF_WDT  (OFF_EF  + 2640)         // 64*260  = 16640
#define OFF_WCL  (OFF_WDT + 16640)        // 64*132  = 8448
#define OFF_PE   (OFF_WCL + 8448)         // 20*68   = 1360
#define OFF_AGG  (OFF_PE  + 1360)         // 5*68    = 340
#define OFF_ATT  (OFF_AGG + 340)          // 5*256   = 1280
#define SMEM_FLOATS (OFF_ATT + 1280)      // 36604 -> pad to 36608
#define SMEM_ALLOC  36608                 // floats (146432 B < 320 KB/WGP)

__device__ __forceinline__ float silu_f(float v)    { return v / (1.0f + __expf(-v)); }
__device__ __forceinline__ float sigmoid_f(float v) { return 1.0f / (1.0f + __expf(-v)); }

__global__ __launch_bounds__(256) void fused_groupattn_kernel(
    const float* __restrict__ X,    // (B*17, 256)
    const float* __restrict__ Wd,   // (256, 64)
    const float* __restrict__ bd,   // (64)
    const float* __restrict__ bn1s, const float* __restrict__ bn1b,
    const float* __restrict__ bn1m, const float* __restrict__ bn1v,
    const float* __restrict__ Wc,   // (64, 128)
    const float* __restrict__ bn2s, const float* __restrict__ bn2b,
    const float* __restrict__ bn2m, const float* __restrict__ bn2v,
    const float* __restrict__ Wa,   // (64, 256)
    const float* __restrict__ ba,   // (256)
    float* __restrict__ Out)
{
    extern __shared__ float smem[];
    float* xs  = smem + OFF_XS;
    float* xd  = smem + OFF_XD;
    float* mns = smem + OFF_MNS;
    float* ef  = smem + OFF_EF;
    float* wdt = smem + OFF_WDT;
    float* wcl = smem + OFF_WCL;
    float* pe  = smem + OFF_PE;
    float* agg = smem + OFF_AGG;
    float* att = smem + OFF_ATT;

    const int b       = blockIdx.x;
    const int tid     = threadIdx.x;
    const int lane    = tid & 31;
    const int w       = tid >> 5;        // wave 0..7
    const int halfsel = lane >> 4;       // K-pair select per WMMA f32 layout
    const int hs2     = halfsel << 1;    // +0 or +2
    const int l16     = lane & 15;

    const float* xg = X + (size_t)b * (K_SLOTS * C_DIM);

    // ---- Phase 0: stage x (non-temporal b128), W_down^T, W_conv into LDS
    for (int i4 = tid; i4 < K_SLOTS * C_DIM / 4; i4 += 256) {     // 1088
        v4f xv = __builtin_nontemporal_load((const v4f*)xg + i4);
        const int k = i4 >> 6, c4 = (i4 & 63) << 2;
        *(v4f*)(xs + k * XS_P + c4) = xv;
    }
    for (int i4 = tid; i4 < C_DIM * CI / 4; i4 += 256) {          // 4096: Wd (256x64) -> wdt[n][k]
        v4f wv = *((const v4f*)Wd + i4);
        const int k = i4 >> 4, n4 = (i4 & 15) << 2;
        wdt[(n4 + 0) * WDT_P + k] = wv.x;
        wdt[(n4 + 1) * WDT_P + k] = wv.y;
        wdt[(n4 + 2) * WDT_P + k] = wv.z;
        wdt[(n4 + 3) * WDT_P + k] = wv.w;
    }
    for (int i4 = tid; i4 < CI * EF_K / 4; i4 += 256) {           // 2048: Wc (64x128) natural copy
        v4f wv = *((const v4f*)Wc + i4);
        const int r = i4 >> 5, c4 = (i4 & 31) << 2;
        *(v4f*)(wcl + r * WCL_P + c4) = wv;
    }
    __syncthreads();

    // ---- Phase 1: xd = silu(bn1(xs @ Wd + bd))   [17x256]@[256x64]
    {
        const int mt = w >> 2, nt = w & 3;
        const int mrow = mt * 16 + l16;                 // rows >=17 read junk, discarded
        const int ncol = nt * 16 + l16;
        const float* aRow = xs  + mrow * XS_P  + hs2;
        const float* bRow = wdt + ncol * WDT_P + hs2;
        v8f acc = {};
        v2f a  = *(const v2f*)(aRow);
        v2f bv = *(const v2f*)(bRow);
#pragma unroll
        for (int kk = 0; kk < C_DIM / 4; ++kk) {
            const int kn = ((kk < C_DIM / 4 - 1) ? (kk + 1) : kk) << 2;
            v2f a2 = *(const v2f*)(aRow + kn);
            v2f b2 = *(const v2f*)(bRow + kn);
            acc = __builtin_amdgcn_wmma_f32_16x16x4_f32(false, a, false, bv,
                                                        (short)0, acc, false, false);
            a = a2; bv = b2;
        }
        const float bdv = bd[ncol];
        const float k1  = bn1s[ncol] * rsqrtf(bn1v[ncol] + 1e-5f);
        const float mu  = bn1m[ncol], bi = bn1b[ncol];
#pragma unroll
        for (int r = 0; r < 8; ++r) {
            const int grow = mt * 16 + r + 8 * halfsel;
            if (grow < K_SLOTS) {
                float v = (acc[r] + bdv - mu) * k1 + bi;
                xd[grow * XD_P + ncol] = silu_f(v);
            }
        }
    }
    __syncthreads();

    // ---- Phase 2: group means (5x64); groups {0..4},{5,7,9},{6,8,10},{11,13,15},{12,14,16}
    for (int i = tid; i < 5 * CI; i += 256) {
        const int g = i >> 6, c = i & 63;
        float s;
        if (g == 0) {
            s = (xd[0*XD_P+c] + xd[1*XD_P+c] + xd[2*XD_P+c] + xd[3*XD_P+c] + xd[4*XD_P+c]) * 0.2f;
        } else {
            const int base = (g <= 2) ? (4 + g) : (8 + g);   // 5,6,11,12
            s = (xd[base*XD_P+c] + xd[(base+2)*XD_P+c] + xd[(base+4)*XD_P+c]) * (1.0f / 3.0f);
        }
        mns[i] = s;
    }
    __syncthreads();

    // ---- Phase 3: ef (20x128): [x_i | x_j - x_i], ROW[e]=e>>2
    for (int i = tid; i < NEDGE * EF_K; i += 256) {
        const int e = i >> 7, c = i & 127;
        const int ri = e >> 2;
        const int j  = e & 3;
        const int ci = j + (j >= ri ? 1 : 0);
        float v;
        if (c < CI) v = mns[ri * CI + c];
        else        v = mns[ci * CI + (c - CI)] - mns[ri * CI + (c - CI)];
        ef[e * EF_P + c] = v;
    }
    __syncthreads();

    // ---- Phase 4: pe = silu(bn2(ef @ Wc^T))   [20x128]@[128x64]
    {
        const int mt = w >> 2, nt = w & 3;
        const int mrow = mt * 16 + l16;
        const int ncol = nt * 16 + l16;
        const float* aRow = ef  + mrow * EF_P  + hs2;
        const float* bRow = wcl + ncol * WCL_P + hs2;
        v8f acc = {};
        v2f a  = *(const v2f*)(aRow);
        v2f bv = *(const v2f*)(bRow);
#pragma unroll
        for (int kk = 0; kk < EF_K / 4; ++kk) {
            const int kn = ((kk < EF_K / 4 - 1) ? (kk + 1) : kk) << 2;
            v2f a2 = *(const v2f*)(aRow + kn);
            v2f b2 = *(const v2f*)(bRow + kn);
            acc = __builtin_amdgcn_wmma_f32_16x16x4_f32(false, a, false, bv,
                                                        (short)0, acc, false, false);
            a = a2; bv = b2;
        }
        const float k1 = bn2s[ncol] * rsqrtf(bn2v[ncol] + 1e-5f);
        const float mu = bn2m[ncol], bi = bn2b[ncol];
#pragma unroll
        for (int r = 0; r < 8; ++r) {
            const int grow = mt * 16 + r + 8 * halfsel;
            if (grow < NEDGE) {
                float v = (acc[r] - mu) * k1 + bi;
                pe[grow * PE_P + ncol] = silu_f(v);
            }
        }
    }
    __syncthreads();

    // ---- Phase 5: agg (5x64) — group g sums edges 4g..4g+3
    for (int i = tid; i < 5 * CI; i += 256) {
        const int g = i >> 6, c = i & 63;
        const int e0 = g * 4;
        agg[g * AGG_P + c] = pe[e0*PE_P+c] + pe[(e0+1)*PE_P+c]
                           + pe[(e0+2)*PE_P+c] + pe[(e0+3)*PE_P+c];
    }
    __syncthreads();

    // ---- Phase 6: att = sigmoid(agg @ Wa + ba)   [5x64]@[64x256]
#pragma unroll
    for (int it = 0; it < 2; ++it) {
        const int nt = w * 2 + it;                       // 0..15
        const int ncol = nt * 16 + l16;
        const float* aRow = agg + l16 * AGG_P + hs2;     // rows >=5 junk, discarded
        v8f acc = {};
        v2f a = *(const v2f*)(aRow);
        v2f bv;
        bv.x = Wa[(hs2 + 0) * C_DIM + ncol];
        bv.y = Wa[(hs2 + 1) * C_DIM + ncol];
#pragma unroll
        for (int kk = 0; kk < CI / 4; ++kk) {
            const int kn = ((kk < CI / 4 - 1) ? (kk + 1) : kk) << 2;
            v2f a2 = *(const v2f*)(aRow + kn);
            v2f b2;
            b2.x = Wa[(kn + hs2 + 0) * C_DIM + ncol];
            b2.y = Wa[(kn + hs2 + 1) * C_DIM + ncol];
            acc = __builtin_amdgcn_wmma_f32_16x16x4_f32(false, a, false, bv,
                                                        (short)0, acc, false, false);
            a = a2; bv = b2;
        }
        const float bav = ba[ncol];
#pragma unroll
        for (int r = 0; r < 8; ++r) {
            const int grow = r + 8 * halfsel;
            if (grow < 5)
                att[grow * C_DIM + ncol] = sigmoid_f(acc[r] + bav);
        }
    }
    __syncthreads();

    // ---- Phase 7: out = x * att[group(k)], b128 non-temporal
    float* og = Out + (size_t)b * (K_SLOTS * C_DIM);
    for (int i4 = tid; i4 < K_SLOTS * C_DIM / 4; i4 += 256) {     // 1088
        const int k = i4 >> 6, c4 = (i4 & 63) << 2;
        const int g = (k < 5) ? 0 : ((k < 11) ? ((k & 1) ? 1 : 2)
                                              : ((k & 1) ? 3 : 4));
        v4f xv = *(const v4f*)(xs + k * XS_P + c4);
        v4f av = *(const v4f*)(att + g * C_DIM + c4);
        __builtin_nontemporal_store(xv * av, (v4f*)og + i4);
    }
}

extern "C" void kernel_launch(void* const* d_in, const int* in_sizes, int n_in,
                              void* d_out, int out_size, void* d_ws, size_t ws_size,
                              hipStream_t stream) {
    const float* X    = (const float*)d_in[0];
    const float* Wd   = (const float*)d_in[1];
    const float* bd   = (const float*)d_in[2];
    const float* bn1s = (const float*)d_in[3];
    const float* bn1b = (const float*)d_in[4];
    const float* bn1m = (const float*)d_in[5];
    const float* bn1v = (const float*)d_in[6];
    const float* Wc   = (const float*)d_in[7];
    const float* bn2s = (const float*)d_in[8];
    const float* bn2b = (const float*)d_in[9];
    const float* bn2m = (const float*)d_in[10];
    const float* bn2v = (const float*)d_in[11];
    const float* Wa   = (const float*)d_in[12];
    const float* ba   = (const float*)d_in[13];
    float* Out = (float*)d_out;

    const int BK = in_sizes[0] / C_DIM;     // B*K
    const int B  = BK / K_SLOTS;

    const size_t shmem = SMEM_ALLOC * sizeof(float);  // 146432 B < 320 KB/WGP
    fused_groupattn_kernel<<<dim3(B), dim3(256), shmem, stream>>>(
        X, Wd, bd, bn1s, bn1b, bn1m, bn1v,
        Wc, bn2s, bn2b, bn2m, bn2v, Wa, ba, Out);
}